// DGLGNN_node_35914516529736
// MI455X (gfx1250) — compile-verified
//
#include <hip/hip_runtime.h>
#include <hip/hip_bf16.h>

// ---------------------------------------------------------------------------
// DGLGNN forward on gfx1250 (MI455X).
//  - Dense node GEMMs ([N,64]x[64,64], f32) via V_WMMA_F32_16X16X4_F32.
//  - Edge message passing via global_atomic_add_f32 into a single accumulator.
//  - LayerNorm via wave32 shuffle reductions.
// ---------------------------------------------------------------------------

#define NI_NODES 60000
#define NN_NODES 40000
#define N_NODES  100000
#define N_EDGES  1000000
#define DIM      64
#define N_LAYERS 4

typedef float v2f __attribute__((ext_vector_type(2)));
typedef float v8f __attribute__((ext_vector_type(8)));

__device__ __forceinline__ float lrelu(float v) {
    return v > 0.0f ? v : 0.1f * v;
}

// ---------------------------------------------------------------------------
// Instance-node encoder: Linear(11,128)+LReLU -> Linear(128,64)+LReLU
// One 64-thread block per node; t1 staged in LDS.
// ---------------------------------------------------------------------------
__global__ void enc_instance_kernel(const float* __restrict__ x,
                                    const float* __restrict__ W1,
                                    const float* __restrict__ b1,
                                    const float* __restrict__ W2,
                                    const float* __restrict__ b2,
                                    float* __restrict__ h0) {
    __shared__ float t1[128];
    const int n = blockIdx.x;
    const int j = threadIdx.x;   // 0..63
    const float* xr = x + (size_t)n * 11;
    float xv[11];
#pragma unroll
    for (int k = 0; k < 11; ++k) xv[k] = xr[k];

#pragma unroll
    for (int jj = 0; jj < 2; ++jj) {
        const int col = j + jj * 64;
        float s = b1[col];
#pragma unroll
        for (int k = 0; k < 11; ++k) s += xv[k] * W1[k * 128 + col];
        t1[col] = lrelu(s);
    }
    __syncthreads();

    float s = b2[j];
#pragma unroll 8
    for (int k = 0; k < 128; ++k) s += t1[k] * W2[k * 64 + j];
    h0[(size_t)n * DIM + j] = lrelu(s);
}

// ---------------------------------------------------------------------------
// Net-node encoder: Linear(3,64)+LReLU -> Linear(64,64)+LReLU
// Also writes level-0 slab of the JK-concat output.
// ---------------------------------------------------------------------------
__global__ void enc_net_kernel(const float* __restrict__ xn,
                               const float* __restrict__ W1,
                               const float* __restrict__ b1,
                               const float* __restrict__ W2,
                               const float* __restrict__ b2,
                               float* __restrict__ h0,
                               float* __restrict__ out) {
    __shared__ float t1[64];
    const int n = blockIdx.x;        // 0..NN-1
    const int j = threadIdx.x;       // 0..63
    const float* xr = xn + (size_t)n * 3;
    float s = b1[j];
#pragma unroll
    for (int k = 0; k < 3; ++k) s += xr[k] * W1[k * 64 + j];
    t1[j] = lrelu(s);
    __syncthreads();

    float s2 = b2[j];
#pragma unroll 8
    for (int k = 0; k < 64; ++k) s2 += t1[k] * W2[k * 64 + j];
    const float v = lrelu(s2);
    h0[(size_t)(NI_NODES + n) * DIM + j] = v;
    out[(size_t)n * (DIM * (N_LAYERS + 1)) + j] = v;   // level 0
}

// ---------------------------------------------------------------------------
// Degree init / accumulate / inverse-sqrt
// ---------------------------------------------------------------------------
__global__ void deg_init_kernel(float* __restrict__ degf, float* __restrict__ degr) {
    const int i = blockIdx.x * blockDim.x + threadIdx.x;
    if (i < N_NODES) { degf[i] = 1.0f; degr[i] = 1.0f; }
}

__global__ void deg_accum_kernel(const int* __restrict__ src,
                                 const int* __restrict__ dst,
                                 float* __restrict__ degf,
                                 float* __restrict__ degr) {
    const int e = blockIdx.x * blockDim.x + threadIdx.x;
    if (e < N_EDGES) {
        atomicAdd(&degf[dst[e]], 1.0f);
        atomicAdd(&degr[src[e]], 1.0f);
    }
}

__global__ void deg_rsqrt_kernel(const float* __restrict__ degf,
                                 const float* __restrict__ degr,
                                 float* __restrict__ disf,
                                 float* __restrict__ disr) {
    const int i = blockIdx.x * blockDim.x + threadIdx.x;
    if (i < N_NODES) {
        disf[i] = rsqrtf(degf[i]);
        disr[i] = rsqrtf(degr[i]);
    }
}

// ---------------------------------------------------------------------------
// Fused per-layer dense kernel:
//   xf = h @ Wf + bf ; xr = h @ Wr + br      (both via V_WMMA_F32_16X16X4_F32)
//   pf  = disf * relu(xf)        (message payload, fwd)
//   pr  = disr * relu(xr)        (message payload, rev)
//   acc = relu(xf+rootf)/degf + relu(xr+rootr)/degr   (both self-loops)
//
// 256 threads = 8 waves; each wave owns one 16x16 output tile; block covers
// 32 rows x 64 cols. N_NODES % 32 == 0 -> no bounds branches (EXEC all-1s,
// as WMMA requires).
//
// Fragment layouts (ISA 7.12.2, 32-bit):
//   A 16x4 : row = row0 + (lane&15); VGPR0/1 hold K = k+2*(lane>>4) (+1)
//   B 4x16 : col = n0  + (lane&15); VGPR0/1 hold K = k+2*(lane>>4) (+1)
//   C/D    : VGPR i -> row = row0 + i + 8*(lane>>4), col = n0 + (lane&15)
// ---------------------------------------------------------------------------
__global__ void gcn_dense_wmma_kernel(const float* __restrict__ h,
                                      const float* __restrict__ Wf,
                                      const float* __restrict__ bf,
                                      const float* __restrict__ rootf,
                                      const float* __restrict__ Wr,
                                      const float* __restrict__ br,
                                      const float* __restrict__ rootr,
                                      const float* __restrict__ degf,
                                      const float* __restrict__ degr,
                                      const float* __restrict__ disf,
                                      const float* __restrict__ disr,
                                      float* __restrict__ pf,
                                      float* __restrict__ pr,
                                      float* __restrict__ acc) {
    const int tid  = threadIdx.x;
    const int wave = tid >> 5;          // 0..7
    const int lane = tid & 31;
    const int half = lane >> 4;         // 0 or 1
    const int lrow = lane & 15;

    const int row0 = (blockIdx.x * 2 + (wave >> 2)) * 16;  // 16-row tile
    const int n0   = (wave & 3) * 16;                      // 16-col tile

    v8f cf = {0.f, 0.f, 0.f, 0.f, 0.f, 0.f, 0.f, 0.f};
    v8f cr = {0.f, 0.f, 0.f, 0.f, 0.f, 0.f, 0.f, 0.f};

    const float* hrow = h + (size_t)(row0 + lrow) * DIM;

#pragma unroll
    for (int k = 0; k < DIM; k += 4) {
        const int kk = k + half * 2;
        // A fragment: two consecutive K values of this lane's row (8B aligned)
        v2f a = *(const v2f*)(hrow + kk);
        // B fragments: W[kk][n0+lrow], W[kk+1][n0+lrow]
        v2f bfv; bfv.x = Wf[kk * DIM + n0 + lrow]; bfv.y = Wf[(kk + 1) * DIM + n0 + lrow];
        v2f brv; brv.x = Wr[kk * DIM + n0 + lrow]; brv.y = Wr[(kk + 1) * DIM + n0 + lrow];

        cf = __builtin_amdgcn_wmma_f32_16x16x4_f32(false, a, false, bfv,
                                                   (short)0, cf, false, false);
        cr = __builtin_amdgcn_wmma_f32_16x16x4_f32(false, a, false, brv,
                                                   (short)0, cr, false, false);
    }

    // Epilogue: bias, relu, dis-scale, self-loop; per-lane column is fixed.
    const int c = n0 + lrow;
    const float bfc = bf[c], brc = br[c];
    const float rfc = rootf[c], rrc = rootr[c];

#pragma unroll
    for (int i = 0; i < 8; ++i) {
        const int r = row0 + half * 8 + i;
        const size_t idx = (size_t)r * DIM + c;
        const float xf = cf[i] + bfc;
        const float xr = cr[i] + brc;
        pf[idx] = disf[r] * fmaxf(xf, 0.0f);
        pr[idx] = disr[r] * fmaxf(xr, 0.0f);
        acc[idx] = fmaxf(xf + rfc, 0.0f) / degf[r]
                 + fmaxf(xr + rrc, 0.0f) / degr[r];
    }
}

// ---------------------------------------------------------------------------
// Edge scatter: acc[dst] += disf[dst]*pf[src]; acc[src] += disr[src]*pr[dst]
// 64 threads per edge (one per channel), 4 edges per 256-thread block.
// ---------------------------------------------------------------------------
__global__ void edge_scatter_kernel(const int* __restrict__ src,
                                    const int* __restrict__ dst,
                                    const float* __restrict__ disf,
                                    const float* __restrict__ disr,
                                    const float* __restrict__ pf,
                                    const float* __restrict__ pr,
                                    float* __restrict__ acc) {
    const int e = blockIdx.x * 4 + (threadIdx.x >> 6);
    const int c = threadIdx.x & 63;
    if (e < N_EDGES) {
        const int s = src[e];
        const int d = dst[e];
        const float vf = disf[d] * pf[(size_t)s * DIM + c];
        atomicAdd(&acc[(size_t)d * DIM + c], vf);
        const float vr = disr[s] * pr[(size_t)d * DIM + c];
        atomicAdd(&acc[(size_t)s * DIM + c], vr);
    }
}

// ---------------------------------------------------------------------------
// Finalize: LayerNorm(acc) * g + b -> LReLU -> + h_cur ; write h_next and
// the (layer+1) slab of the JK-concat output for net nodes.
// One wave32 per node (2 channels/lane), 8 nodes per block.
// ---------------------------------------------------------------------------
__global__ void layer_finalize_kernel(const float* __restrict__ acc,
                                      const float* __restrict__ h_cur,
                                      const float* __restrict__ g,
                                      const float* __restrict__ b,
                                      float* __restrict__ h_next,
                                      float* __restrict__ out,
                                      int layer) {
    const int wave = threadIdx.x >> 5;
    const int lane = threadIdx.x & 31;
    const int n = blockIdx.x * 8 + wave;   // N_NODES % 8 == 0

    const size_t base = (size_t)n * DIM;
    const float a0 = acc[base + lane];
    const float a1 = acc[base + lane + 32];

    float s  = a0 + a1;
    float sq = a0 * a0 + a1 * a1;
#pragma unroll
    for (int off = 16; off > 0; off >>= 1) {
        s  += __shfl_xor(s,  off, 32);
        sq += __shfl_xor(sq, off, 32);
    }
    const float mean = s * (1.0f / DIM);
    const float var  = sq * (1.0f / DIM) - mean * mean;
    const float inv  = rsqrtf(var + 1e-5f);

    const int c0 = lane, c1 = lane + 32;
    float v0 = lrelu((a0 - mean) * inv * g[c0] + b[c0]) + h_cur[base + c0];
    float v1 = lrelu((a1 - mean) * inv * g[c1] + b[c1]) + h_cur[base + c1];
    h_next[base + c0] = v0;
    h_next[base + c1] = v1;

    if (n >= NI_NODES) {
        const size_t ob = (size_t)(n - NI_NODES) * (DIM * (N_LAYERS + 1))
                        + (size_t)(layer + 1) * DIM;
        out[ob + c0] = v0;
        out[ob + c1] = v1;
    }
}

// ---------------------------------------------------------------------------
// Host-side launcher
// ---------------------------------------------------------------------------
extern "C" void kernel_launch(void* const* d_in, const int* in_sizes, int n_in,
                              void* d_out, int out_size, void* d_ws, size_t ws_size,
                              hipStream_t stream) {
    (void)in_sizes; (void)n_in; (void)out_size; (void)ws_size;

    const float* x       = (const float*)d_in[0];
    const float* x_net   = (const float*)d_in[1];
    const int*   src     = (const int*)d_in[2];
    const int*   dst     = (const int*)d_in[3];
    // d_in[4]: num_instances (known constant NI_NODES)
    const float* enc_W1  = (const float*)d_in[5];
    const float* enc_b1  = (const float*)d_in[6];
    const float* enc_W2  = (const float*)d_in[7];
    const float* enc_b2  = (const float*)d_in[8];
    const float* net_W1  = (const float*)d_in[9];
    const float* net_b1  = (const float*)d_in[10];
    const float* net_W2  = (const float*)d_in[11];
    const float* net_b2  = (const float*)d_in[12];
    const float* conv_W  = (const float*)d_in[13];
    const float* conv_b  = (const float*)d_in[14];
    const float* conv_rt = (const float*)d_in[15];
    const float* rconv_W = (const float*)d_in[16];
    const float* rconv_b = (const float*)d_in[17];
    const float* rconv_rt= (const float*)d_in[18];
    const float* ln_g    = (const float*)d_in[19];
    const float* ln_b    = (const float*)d_in[20];

    float* out = (float*)d_out;

    // Workspace carve-out (~130 MB)
    float* ws = (float*)d_ws;
    const size_t ND = (size_t)N_NODES * DIM;
    float* h0   = ws;
    float* h1   = ws + ND;
    float* pf   = ws + 2 * ND;
    float* pr   = ws + 3 * ND;
    float* acc  = ws + 4 * ND;
    float* degf = ws + 5 * ND;
    float* degr = degf + N_NODES;
    float* disf = degr + N_NODES;
    float* disr = disf + N_NODES;

    // 1) Encoders -> h0 (and level-0 output slab for net nodes)
    enc_instance_kernel<<<NI_NODES, 64, 0, stream>>>(x, enc_W1, enc_b1, enc_W2, enc_b2, h0);
    enc_net_kernel<<<NN_NODES, 64, 0, stream>>>(x_net, net_W1, net_b1, net_W2, net_b2, h0, out);

    // 2) Degrees and normalizers
    deg_init_kernel<<<(N_NODES + 255) / 256, 256, 0, stream>>>(degf, degr);
    deg_accum_kernel<<<(N_EDGES + 255) / 256, 256, 0, stream>>>(src, dst, degf, degr);
    deg_rsqrt_kernel<<<(N_NODES + 255) / 256, 256, 0, stream>>>(degf, degr, disf, disr);

    // 3) Layers
    float* h_cur = h0;
    float* h_next = h1;
    for (int l = 0; l < N_LAYERS; ++l) {
        const float* Wf = conv_W  + (size_t)l * DIM * DIM;
        const float* Wr = rconv_W + (size_t)l * DIM * DIM;
        gcn_dense_wmma_kernel<<<N_NODES / 32, 256, 0, stream>>>(
            h_cur, Wf, conv_b + l * DIM, conv_rt + l * DIM,
            Wr, rconv_b + l * DIM, rconv_rt + l * DIM,
            degf, degr, disf, disr, pf, pr, acc);

        edge_scatter_kernel<<<N_EDGES / 4, 256, 0, stream>>>(
            src, dst, disf, disr, pf, pr, acc);

        layer_finalize_kernel<<<N_NODES / 8, 256, 0, stream>>>(
            acc, h_cur, ln_g + l * DIM, ln_b + l * DIM, h_next, out, l);

        float* t = h_cur; h_cur = h_next; h_next = t;
    }
}